// CustomRBF_74234214744185
// MI455X (gfx1250) — compile-verified
//
#include <hip/hip_runtime.h>
#include <hip/hip_bf16.h>

typedef __attribute__((ext_vector_type(2))) float v2f;
typedef __attribute__((ext_vector_type(8))) float v8f;

#define D_FEAT 128
#define GAMMA (1.0f / 128.0f)

// One wave32 handles 16-row tiles of X. For each tile:
//   A (16x4 f32 WMMA tile) = (x - c)^2 slices, B = ones  =>  D row-sums.
// 32 WMMA k-steps cover all 128 features; accumulator stays f32 throughout.
__global__ __launch_bounds__(256) void rbf_wmma_kernel(
    const float* __restrict__ X,
    const float* __restrict__ centroid,
    const float* __restrict__ w,
    const float* __restrict__ b,
    float* __restrict__ out,
    int n, int ntiles, int totalWaves)
{
  const int lane       = threadIdx.x & 31;
  const int waveInBlk  = threadIdx.x >> 5;
  const int gwave      = blockIdx.x * (blockDim.x >> 5) + waveInBlk;

  const int rowInTile  = lane & 15;   // M index within the 16-row tile
  const int hi         = lane >> 4;   // 0: lanes 0-15 (K+0,K+1), 1: lanes 16-31 (K+2,K+3)

  const float w0 = w[0];
  const float b0 = b[0];

  // Preload the centroid slices this lane needs: step s uses c[4s + 2*hi .. +1].
  const v2f* cent2 = (const v2f*)centroid;
  v2f c2[32];
#pragma unroll
  for (int s = 0; s < 32; ++s) c2[s] = cent2[2 * s + hi];

  v2f ones; ones.x = 1.0f; ones.y = 1.0f;

  for (int t = gwave; t < ntiles; t += totalWaves) {
    const int row0 = t * 16;
    int row = row0 + rowInTile;
    if (row >= n) row = n - 1;                 // tail clamp (affects loads only)
    const v2f* xr2 = (const v2f*)(X + (size_t)row * D_FEAT);

    v8f acc0 = {};
    v8f acc1 = {};
#pragma unroll
    for (int s = 0; s < 32; ++s) {
      v2f xv = xr2[2 * s + hi];                // global_load_b64, imm offset 16*s
      v2f d;  d.x  = xv.x - c2[s].x;  d.y  = xv.y - c2[s].y;
      v2f d2; d2.x = d.x * d.x;       d2.y = d.y * d.y;
      if (s & 1)
        acc1 = __builtin_amdgcn_wmma_f32_16x16x4_f32(
                   false, d2, false, ones, (short)0, acc1, false, false);
      else
        acc0 = __builtin_amdgcn_wmma_f32_16x16x4_f32(
                   false, d2, false, ones, (short)0, acc0, false, false);
    }

    // C/D layout: VGPR v holds M=v (lanes 0-15) / M=v+8 (lanes 16-31), N=lane%16.
    // All N columns are identical (B was ones), so lane 0 owns rows row0+0..7 and
    // lane 16 owns rows row0+8..15 in acc[0..7].
    if (rowInTile == 0) {
      const int mbase = row0 + (hi << 3);
      float r[8];
#pragma unroll
      for (int i = 0; i < 8; ++i) {
        float sq = acc0[i] + acc1[i];
        r[i] = fmaf(w0, __expf(-GAMMA * sq), b0);
      }
      if (mbase + 8 <= n) {
        *(float4*)(out + mbase)     = make_float4(r[0], r[1], r[2], r[3]);
        *(float4*)(out + mbase + 4) = make_float4(r[4], r[5], r[6], r[7]);
      } else {
#pragma unroll
        for (int i = 0; i < 8; ++i)
          if (mbase + i < n) out[mbase + i] = r[i];
      }
    }
  }
}

extern "C" void kernel_launch(void* const* d_in, const int* in_sizes, int n_in,
                              void* d_out, int out_size, void* d_ws, size_t ws_size,
                              hipStream_t stream) {
  const float* X        = (const float*)d_in[0];
  const float* centroid = (const float*)d_in[1];
  const float* w        = (const float*)d_in[2];
  const float* b        = (const float*)d_in[3];
  float* out            = (float*)d_out;

  const int n      = in_sizes[0] / D_FEAT;       // 1,000,000 rows
  const int ntiles = (n + 15) / 16;              // 62,500 tiles
  const int wavesPerBlock = 256 / 32;            // 8 waves (wave32)

  int blocks = (ntiles + wavesPerBlock - 1) / wavesPerBlock;
  if (blocks > 2048) blocks = 2048;              // grid-stride: ~4 tiles/wave,
                                                 // amortizes centroid preload
  const int totalWaves = blocks * wavesPerBlock;

  rbf_wmma_kernel<<<blocks, 256, 0, stream>>>(X, centroid, w, b, out,
                                              n, ntiles, totalWaves);
}